// EvenNet_70188355551843
// MI455X (gfx1250) — compile-verified
//
#include <hip/hip_runtime.h>
#include <hip/hip_bf16.h>

#define N_NODES 100000
#define N_EDGES 1600000
#define C_IN    500
#define HIDDEN  256
#define C_OUT   50
#define K_HALF  5

typedef __attribute__((ext_vector_type(2))) float v2f;
typedef __attribute__((ext_vector_type(8))) float v8f;

// ---------------------------------------------------------------- degree / norm
__global__ void deg_kernel(const int* __restrict__ src, float* __restrict__ deg) {
    int e = blockIdx.x * blockDim.x + threadIdx.x;
    if (e < N_EDGES) atomicAdd(&deg[src[e]], 1.0f);
}

__global__ void dinv_kernel(float* __restrict__ deg) {   // in-place deg -> d^-1/2
    int i = blockIdx.x * blockDim.x + threadIdx.x;
    if (i < N_NODES) {
        float d = deg[i];
        deg[i] = (d > 0.0f) ? rsqrtf(d) : 0.0f;
    }
}

__global__ void norm_kernel(const int* __restrict__ src, const int* __restrict__ dst,
                            const float* __restrict__ dinv, float* __restrict__ norm) {
    int e = blockIdx.x * blockDim.x + threadIdx.x;
    if (e < N_EDGES) norm[e] = dinv[src[e]] * dinv[dst[e]];
}

// ---------------------------------------------------------------- fused MLP
// One block = one 16-row strip. Phase 1: H = relu(X@W1+b1) via 16 waves of
// fp32 WMMA (K=500). Phase 2: out = H@W2+b2 via 4 waves (K=256). H lives
// only in LDS (saves 2x100MB of HBM traffic).
#define HS_LD 260   // 256+4: stride%64==4 -> conflict-free K-major reads

__launch_bounds__(512)
__global__ void mlp_kernel(const float* __restrict__ X,  const float* __restrict__ W1,
                           const float* __restrict__ b1, const float* __restrict__ W2,
                           const float* __restrict__ b2, const float* __restrict__ temp,
                           float* __restrict__ hid, float* __restrict__ xk) {
    __shared__ float Xs[16 * C_IN];    // 32 KB
    __shared__ float Hs[16 * HS_LD];   // ~16.6 KB

    const int tid = threadIdx.x;
    const int r0  = blockIdx.x * 16;

    // X strip is contiguous in row-major global memory: linear coalesced copy
    for (int t = tid; t < 16 * C_IN; t += 512) Xs[t] = X[r0 * C_IN + t];
    __syncthreads();

    const int lane   = tid & 31;
    const int wave   = tid >> 5;
    const int n      = lane & 15;          // A-row (M) / B-col (N) index
    const int kk     = (lane >> 4) * 2;    // K sub-offset per half-wave
    const int rowoff = (lane >> 4) * 8;    // C/D row offset per half-wave

    // ---- phase 1: each wave one 16x16 tile of H (cols c0..c0+15)
    {
        const int c0 = wave * 16;
        const float bias = b1[c0 + n];
        v8f acc;
        #pragma unroll
        for (int i = 0; i < 8; ++i) acc[i] = bias;

        for (int k0 = 0; k0 < C_IN; k0 += 4) {
            v2f a, b;
            a.x = Xs[n * C_IN + k0 + kk];
            a.y = Xs[n * C_IN + k0 + kk + 1];
            b.x = W1[(k0 + kk) * HIDDEN + c0 + n];
            b.y = W1[(k0 + kk + 1) * HIDDEN + c0 + n];
            acc = __builtin_amdgcn_wmma_f32_16x16x4_f32(
                false, a, false, b, (short)0, acc, false, false);
        }
        #pragma unroll
        for (int i = 0; i < 8; ++i) {
            float v = acc[i] > 0.0f ? acc[i] : 0.0f;     // relu
            Hs[(i + rowoff) * HS_LD + c0 + n] = v;
        }
    }
    __syncthreads();

    // ---- phase 2: waves 0..3 compute output cols [wave*16, wave*16+16)
    if (wave < 4) {
        const int c0  = wave * 16;
        const int col = c0 + n;
        const bool cv = (col < C_OUT);
        const float bias = cv ? b2[col] : 0.0f;
        v8f acc;
        #pragma unroll
        for (int i = 0; i < 8; ++i) acc[i] = bias;

        for (int k0 = 0; k0 < HIDDEN; k0 += 4) {
            v2f a, b;
            a.x = Hs[n * HS_LD + k0 + kk];
            a.y = Hs[n * HS_LD + k0 + kk + 1];
            b.x = cv ? W2[(k0 + kk) * C_OUT + col] : 0.0f;
            b.y = cv ? W2[(k0 + kk + 1) * C_OUT + col] : 0.0f;
            acc = __builtin_amdgcn_wmma_f32_16x16x4_f32(
                false, a, false, b, (short)0, acc, false, false);
        }
        const float t0 = temp[0];
        #pragma unroll
        for (int i = 0; i < 8; ++i) {
            if (cv) {
                int row = r0 + i + rowoff;
                float v = acc[i];
                hid[row * C_OUT + col] = t0 * v;   // hidden0 = temp[0]*h
                xk [row * C_OUT + col] = v;        // xk = h
            }
        }
    }
}

// ---------------------------------------------------------------- propagation
// 64 threads per edge (f = tid&63); edge index forced scalar so src/dst/norm
// become s-loads. Scatter via global_atomic_add_f32; xin/xout are L2-resident.
__global__ void scatter_kernel(const int* __restrict__ src, const int* __restrict__ dst,
                               const float* __restrict__ norm, const float* __restrict__ xin,
                               float* __restrict__ xout) {
    int e = (blockIdx.x << 2) + (threadIdx.x >> 6);   // 256 thr = 4 edges/block
    int f = threadIdx.x & 63;
    e = __builtin_amdgcn_readfirstlane(e);            // uniform per wave
    if (e >= N_EDGES || f >= C_OUT) return;
    float w = norm[e];
    int s = src[e], d = dst[e];
    atomicAdd(&xout[d * C_OUT + f], w * xin[s * C_OUT + f]);
}

__global__ void axpy_kernel(float* __restrict__ hid, const float* __restrict__ xk,
                            const float* __restrict__ temp, int idx) {
    int i = blockIdx.x * blockDim.x + threadIdx.x;
    if (i < N_NODES * C_OUT) hid[i] += temp[idx] * xk[i];
}

// ---------------------------------------------------------------- log_softmax
// wave32-native: one wave per row of 50, shfl_xor tree reductions
__global__ void lsm_kernel(const float* __restrict__ hid, float* __restrict__ out) {
    int row = blockIdx.x * 8 + (threadIdx.x >> 5);
    if (row >= N_NODES) return;
    int lane = threadIdx.x & 31;
    const float* p = hid + row * C_OUT;
    float v1 = (lane < C_OUT)      ? p[lane]      : -__builtin_inff();
    float v2 = (lane + 32 < C_OUT) ? p[lane + 32] : -__builtin_inff();
    float m = fmaxf(v1, v2);
    #pragma unroll
    for (int off = 16; off; off >>= 1) m = fmaxf(m, __shfl_xor(m, off, 32));
    float s = 0.0f;
    if (lane < C_OUT)      s += expf(v1 - m);
    if (lane + 32 < C_OUT) s += expf(v2 - m);
    #pragma unroll
    for (int off = 16; off; off >>= 1) s += __shfl_xor(s, off, 32);
    float lse = m + logf(s);
    if (lane < C_OUT)      out[row * C_OUT + lane]      = v1 - lse;
    if (lane + 32 < C_OUT) out[row * C_OUT + lane + 32] = v2 - lse;
}

// ---------------------------------------------------------------- launch
extern "C" void kernel_launch(void* const* d_in, const int* in_sizes, int n_in,
                              void* d_out, int out_size, void* d_ws, size_t ws_size,
                              hipStream_t stream) {
    const float* x    = (const float*)d_in[0];
    const int*   ei   = (const int*)  d_in[1];
    const float* W1   = (const float*)d_in[2];
    const float* b1   = (const float*)d_in[3];
    const float* W2   = (const float*)d_in[4];
    const float* b2   = (const float*)d_in[5];
    const float* temp = (const float*)d_in[6];
    float* out = (float*)d_out;

    const int* src = ei;              // edge_index[0]
    const int* dst = ei + N_EDGES;    // edge_index[1]

    float* ws   = (float*)d_ws;
    float* deg  = ws;                             // N  (becomes dinv in-place)
    float* norm = deg + N_NODES;                  // E
    float* hid  = norm + N_EDGES;                 // N*C_OUT
    float* xkA  = hid + (size_t)N_NODES * C_OUT;  // N*C_OUT
    float* xkB  = xkA + (size_t)N_NODES * C_OUT;  // N*C_OUT

    const size_t featBytes = (size_t)N_NODES * C_OUT * sizeof(float);

    hipMemsetAsync(deg, 0, N_NODES * sizeof(float), stream);
    deg_kernel <<<(N_EDGES + 255) / 256, 256, 0, stream>>>(src, deg);
    dinv_kernel<<<(N_NODES + 255) / 256, 256, 0, stream>>>(deg);
    norm_kernel<<<(N_EDGES + 255) / 256, 256, 0, stream>>>(src, dst, deg, norm);

    mlp_kernel<<<N_NODES / 16, 512, 0, stream>>>(x, W1, b1, W2, b2, temp, hid, xkA);

    const int scatterGrid = N_EDGES / 4;   // 4 edges per 256-thread block
    for (int k = 0; k < K_HALF; ++k) {
        hipMemsetAsync(xkB, 0, featBytes, stream);
        scatter_kernel<<<scatterGrid, 256, 0, stream>>>(src, dst, norm, xkA, xkB);
        hipMemsetAsync(xkA, 0, featBytes, stream);
        scatter_kernel<<<scatterGrid, 256, 0, stream>>>(src, dst, norm, xkB, xkA);
        axpy_kernel<<<(N_NODES * C_OUT + 255) / 256, 256, 0, stream>>>(hid, xkA, temp, k + 1);
    }

    lsm_kernel<<<(N_NODES + 7) / 8, 256, 0, stream>>>(hid, out);
}